// SelectiveScan_37134287241922
// MI455X (gfx1250) — compile-verified
//
#include <hip/hip_runtime.h>
#include <hip/hip_bf16.h>
#include <math.h>

// Model dims (fixed by the reference)
#define D_MODEL 1024
#define D_STATE 16
#define D_CONV  4
#define D_INNER 2048
#define DT_RANK 64
#define SEQ_L   2048

typedef float v2f __attribute__((ext_vector_type(2)));
typedef float v8f __attribute__((ext_vector_type(8)));

__device__ __forceinline__ float softplusf(float v) {
    return (v > 20.0f) ? v : log1pf(__expf(v));
}
__device__ __forceinline__ float siluf(float v) {
    return v / (1.0f + __expf(-v));
}

// ---------------------------------------------------------------------------
// fp32 WMMA GEMM:  C[M,N] = A[M,K] * W[N,K]^T   (all row-major fp32)
// One wave computes a (16*MT) x (16*NQ) tile of C. K stepped by 4 with
// V_WMMA_F32_16X16X4_F32. Fully-unrolled, branch-free, software-pipelined
// K loop: the k+4 fragment loads are in flight while the current MT*NQ
// WMMAs issue. Requires K % 4 == 0, K >= 8, grid exactly tiles M and N.
// EPI: false = plain store, true = softplus(v + bias[n]) (dt path).
// ---------------------------------------------------------------------------
template <int NQ, int MT, bool EPI>
__global__ __launch_bounds__(32)
void wmma_gemm_nt(const float* __restrict__ A, const float* __restrict__ W,
                  float* __restrict__ C, int K,
                  int lda, int ldw, int ldc,
                  const float* __restrict__ bias)
{
    const int lane = threadIdx.x;        // 0..31, wave32
    const int half = lane >> 4;          // 0: K+0..1, 1: K+2..3
    const int idx  = lane & 15;          // M-row / N-col within 16x16 tile
    const int koff = half * 2;

    const int m0 = blockIdx.y * (16 * MT);
    const int n0 = blockIdx.x * (16 * NQ);

    // Per-lane fragment base pointers (ISA 7.12.2 fp32 A/B layouts)
    const float* ap[MT];
    #pragma unroll
    for (int mt = 0; mt < MT; ++mt)
        ap[mt] = A + (size_t)(m0 + mt * 16 + idx) * lda + koff;
    const float* wp[NQ];
    #pragma unroll
    for (int q = 0; q < NQ; ++q)
        wp[q] = W + (size_t)(n0 + q * 16 + idx) * ldw + koff;

    v8f acc[MT][NQ];
    #pragma unroll
    for (int mt = 0; mt < MT; ++mt)
        #pragma unroll
        for (int q = 0; q < NQ; ++q)
            acc[mt][q] = (v8f){};

    // Prologue: fragments for k = 0
    v2f a_cur[MT], b_cur[NQ];
    #pragma unroll
    for (int mt = 0; mt < MT; ++mt) a_cur[mt] = *(const v2f*)(ap[mt]);
    #pragma unroll
    for (int q = 0; q < NQ; ++q)    b_cur[q]  = *(const v2f*)(wp[q]);

    // Pipelined main loop: prefetch k+4 while issuing WMMAs for k
    for (int k = 0; k < K - 4; k += 4) {
        v2f a_nxt[MT], b_nxt[NQ];
        #pragma unroll
        for (int mt = 0; mt < MT; ++mt) a_nxt[mt] = *(const v2f*)(ap[mt] + k + 4);
        #pragma unroll
        for (int q = 0; q < NQ; ++q)    b_nxt[q]  = *(const v2f*)(wp[q] + k + 4);

        #pragma unroll
        for (int mt = 0; mt < MT; ++mt)
            #pragma unroll
            for (int q = 0; q < NQ; ++q)
                acc[mt][q] = __builtin_amdgcn_wmma_f32_16x16x4_f32(
                    false, a_cur[mt], false, b_cur[q], (short)0,
                    acc[mt][q], false, false);

        #pragma unroll
        for (int mt = 0; mt < MT; ++mt) a_cur[mt] = a_nxt[mt];
        #pragma unroll
        for (int q = 0; q < NQ; ++q)    b_cur[q]  = b_nxt[q];
    }
    // Epilogue WMMAs for the last k-step
    #pragma unroll
    for (int mt = 0; mt < MT; ++mt)
        #pragma unroll
        for (int q = 0; q < NQ; ++q)
            acc[mt][q] = __builtin_amdgcn_wmma_f32_16x16x4_f32(
                false, a_cur[mt], false, b_cur[q], (short)0,
                acc[mt][q], false, false);

    // Store: C/D layout (ISA 7.12.2): VGPR r -> row r + 8*half, col = idx
    #pragma unroll
    for (int mt = 0; mt < MT; ++mt) {
        #pragma unroll
        for (int q = 0; q < NQ; ++q) {
            const int n = n0 + q * 16 + idx;
            #pragma unroll
            for (int r = 0; r < 8; ++r) {
                const int row = m0 + mt * 16 + r + half * 8;
                float v = acc[mt][q][r];
                if (EPI) v = softplusf(v + bias[n]);
                C[(size_t)row * ldc + n] = v;
            }
        }
    }
}

// ---------------------------------------------------------------------------
// Depthwise causal conv (width 4) + bias + SiLU on the xi half of xz.
// xz: [L, 2*D_INNER]; xi_out: [L, D_INNER]
// ---------------------------------------------------------------------------
__global__ __launch_bounds__(256)
void conv_silu_kernel(const float* __restrict__ xz, const float* __restrict__ cw,
                      const float* __restrict__ cb, float* __restrict__ xi)
{
    const int idx = blockIdx.x * blockDim.x + threadIdx.x;   // over L*D_INNER
    if (idx >= SEQ_L * D_INNER) return;
    const int d = idx & (D_INNER - 1);
    const int l = idx >> 11;                                  // D_INNER == 2048

    float acc = cb[d];
    #pragma unroll
    for (int j = 0; j < D_CONV; ++j) {
        const int ls = l - (D_CONV - 1) + j;
        const float xv = (ls >= 0) ? xz[(size_t)ls * (2 * D_INNER) + d] : 0.0f;
        acc = fmaf(xv, cw[d * D_CONV + j], acc);
    }
    xi[idx] = siluf(acc);
}

// ---------------------------------------------------------------------------
// Fused selective scan. One thread per channel d (2048 threads total).
// State h[16] in registers; a/b recomputed on the fly (no L*D*N temporaries).
// B/C rows (broadcast across all channels) staged through LDS, 128 steps/tile.
// Epilogue: y = (h . C) + D*xi, then y *= silu(z); y2 stored over the dead
// xi-half of the xz workspace (stride 4096) so out_proj can stream it.
// ---------------------------------------------------------------------------
#define SCAN_TL 128
__global__ __launch_bounds__(256)
void scan_kernel(const float* __restrict__ dtb, const float* __restrict__ xi,
                 const float* __restrict__ xdbl, const float* __restrict__ A_log,
                 const float* __restrict__ Dp, const float* __restrict__ xz,
                 float* __restrict__ y2)
{
    __shared__ float sBC[SCAN_TL][2 * D_STATE];   // 128 * 32 floats = 16 KB

    const int d = blockIdx.x * blockDim.x + threadIdx.x;   // 0..2047

    float Arow[D_STATE], h[D_STATE];
    #pragma unroll
    for (int n = 0; n < D_STATE; ++n) {
        Arow[n] = -__expf(A_log[d * D_STATE + n]);
        h[n] = 0.0f;
    }
    const float Dd = Dp[d];

    for (int l0 = 0; l0 < SEQ_L; l0 += SCAN_TL) {
        __syncthreads();
        // Cooperative load of B (cols 64..79) and C (cols 80..95) rows
        for (int t = threadIdx.x; t < SCAN_TL * 2 * D_STATE; t += 256) {
            const int ll = t >> 5, c = t & 31;
            sBC[ll][c] = xdbl[(size_t)(l0 + ll) * (DT_RANK + 2 * D_STATE) + DT_RANK + c];
        }
        __syncthreads();

        for (int li = 0; li < SCAN_TL; ++li) {
            const int l = l0 + li;
            const float dt = dtb[(size_t)l * D_INNER + d];
            const float xv = xi[(size_t)l * D_INNER + d];
            const float db = dt * xv;
            float y = Dd * xv;
            #pragma unroll
            for (int n = 0; n < D_STATE; ++n) {
                const float a = __expf(dt * Arow[n]);
                h[n] = fmaf(a, h[n], db * sBC[li][n]);
                y = fmaf(h[n], sBC[li][D_STATE + n], y);
            }
            const float z = xz[(size_t)l * (2 * D_INNER) + D_INNER + d];
            y2[(size_t)l * (2 * D_INNER) + d] = y * siluf(z);
        }
    }
}

// ---------------------------------------------------------------------------
extern "C" void kernel_launch(void* const* d_in, const int* in_sizes, int n_in,
                              void* d_out, int out_size, void* d_ws, size_t ws_size,
                              hipStream_t stream)
{
    const float* x          = (const float*)d_in[0];
    const float* in_proj_w  = (const float*)d_in[1];
    const float* conv_w     = (const float*)d_in[2];
    const float* conv_b     = (const float*)d_in[3];
    const float* A_log      = (const float*)d_in[4];
    const float* x_proj_w   = (const float*)d_in[5];
    const float* dt_proj_w  = (const float*)d_in[6];
    const float* dt_proj_b  = (const float*)d_in[7];
    const float* Dp         = (const float*)d_in[8];
    const float* out_proj_w = (const float*)d_in[9];
    float* out = (float*)d_out;

    // Workspace layout (floats). y2 aliases the xi-half of xz after the conv
    // kernel has consumed it, shrinking scratch to ~65 MB.
    float* ws   = (float*)d_ws;
    float* xz   = ws;                                   // [2048, 4096]  32 MB
    float* xi   = xz + (size_t)SEQ_L * 2 * D_INNER;     // [2048, 2048]  16 MB
    float* xdbl = xi + (size_t)SEQ_L * D_INNER;         // [2048, 96]    0.75 MB
    float* dtb  = xdbl + (size_t)SEQ_L * (DT_RANK + 2 * D_STATE); // [2048, 2048] 16 MB

    const dim3 wave(32, 1, 1);

    // 1) in_proj: xz[L,4096] = x[L,1024] * in_proj_w[4096,1024]^T
    //    32(M) x 64(N) per wave -> grid (4096/64, 2048/32)
    wmma_gemm_nt<4, 2, false><<<dim3(64, 64), wave, 0, stream>>>(
        x, in_proj_w, xz, D_MODEL, D_MODEL, D_MODEL, 2 * D_INNER, nullptr);

    // 2) depthwise conv + bias + SiLU -> xi[L,2048]
    conv_silu_kernel<<<(SEQ_L * D_INNER) / 256, 256, 0, stream>>>(
        xz, conv_w, conv_b, xi);

    // 3) x_proj: xdbl[L,96] = xi[L,2048] * x_proj_w[96,2048]^T
    //    N = 96 = 2 x 48 -> NQ=3, grid (2, 64), no edge guards needed
    wmma_gemm_nt<3, 2, false><<<dim3(2, 64), wave, 0, stream>>>(
        xi, x_proj_w, xdbl, D_INNER, D_INNER, D_INNER,
        DT_RANK + 2 * D_STATE, nullptr);

    // 4) dt: dtb[L,2048] = softplus(xdbl[:, :64] * dt_proj_w[2048,64]^T + b)
    wmma_gemm_nt<4, 2, true><<<dim3(32, 64), wave, 0, stream>>>(
        xdbl, dt_proj_w, dtb, DT_RANK, DT_RANK + 2 * D_STATE, DT_RANK,
        D_INNER, dt_proj_b);

    // 5) fused selective scan (+ D*xi, * silu(z)); y2 -> xi-half of xz
    scan_kernel<<<D_INNER / 256, 256, 0, stream>>>(
        dtb, xi, xdbl, A_log, Dp, xz, xz);

    // 6) out_proj: out[L,1024] = y2[L,2048](stride 4096) * out_proj_w[1024,2048]^T
    wmma_gemm_nt<4, 2, false><<<dim3(16, 64), wave, 0, stream>>>(
        xz, out_proj_w, out, D_INNER, 2 * D_INNER, D_INNER, D_MODEL, nullptr);
}